// SlotAttentionModel_80642305950454
// MI455X (gfx1250) — compile-verified
//
#include <hip/hip_runtime.h>
#include <hip/hip_bf16.h>

typedef float v2f __attribute__((ext_vector_type(2)));
typedef float v8f __attribute__((ext_vector_type(8)));

#define B_ 16
#define N_ 38400
#define D_ 64
#define S_ 7
#define H_ 128
#define LN_EPS 1e-5f
#define EPS_ 1e-8f
#define TPB_ (N_ / 16)   // tiles per batch = 2400
#define TPW_ 8           // tiles per wave
#define WPB_ 4           // waves per block in k_attn (128 threads)
#define GRID2_ 1200      // 1200 blocks * 4 waves * 8 tiles = 38400 tiles

__device__ __forceinline__ float sigmoidf_(float x) { return 1.f / (1.f + __expf(-x)); }

// s_wait_asynccnt 0: wait for all outstanding async global->LDS ops of this wave.
// asm form doubles as a compiler memory-scheduling barrier.
__device__ __forceinline__ void async_wait0() {
  asm volatile("s_wait_asynccnt 0" ::: "memory");
}

// ---------------------------------------------------------------------------
// K0: slots = mu + exp(log_sigma) * slots_init        (7168 elements)
// ---------------------------------------------------------------------------
__global__ void k_init_slots(const float* __restrict__ sinit, const float* __restrict__ mu,
                             const float* __restrict__ lsig, float* __restrict__ slots) {
  int i = blockIdx.x * blockDim.x + threadIdx.x;
  if (i < B_ * S_ * D_) {
    int d = i & 63;
    slots[i] = mu[d] + __expf(lsig[d]) * sinit[i];
  }
}

// ---------------------------------------------------------------------------
// K1 (per iteration): s = LN(slots); q = s@Wq^T; qk = scale*(q@Wk)  [B,7,64]
// Also zeroes the US/CS accumulators for this batch. Tiny (112 rows).
// ---------------------------------------------------------------------------
__global__ __launch_bounds__(256) void k_qk(const float* __restrict__ slots,
                                            const float* __restrict__ g, const float* __restrict__ bb,
                                            const float* __restrict__ Wq, const float* __restrict__ Wk,
                                            float* __restrict__ qk, float* __restrict__ US,
                                            float* __restrict__ CS) {
  __shared__ float sl[S_][D_];
  __shared__ float qs[S_][D_];
  __shared__ float mrs[S_][2];
  int b = blockIdx.x, t = threadIdx.x;
  for (int i = t; i < S_ * D_; i += 256) {
    US[b * S_ * D_ + i] = 0.f;
    sl[i >> 6][i & 63] = slots[b * S_ * D_ + i];
  }
  if (t < S_) CS[b * S_ + t] = 0.f;
  __syncthreads();
  if (t < S_) {
    float s0 = 0.f, s1 = 0.f;
    for (int d = 0; d < D_; ++d) { float x = sl[t][d]; s0 += x; s1 += x * x; }
    float m = s0 * (1.f / 64.f);
    float v = s1 * (1.f / 64.f) - m * m;
    mrs[t][0] = m; mrs[t][1] = rsqrtf(v + LN_EPS);
  }
  __syncthreads();
  for (int i = t; i < S_ * D_; i += 256) {
    int s = i >> 6, d = i & 63;
    sl[s][d] = (sl[s][d] - mrs[s][0]) * mrs[s][1] * g[d] + bb[d];
  }
  __syncthreads();
  for (int i = t; i < S_ * D_; i += 256) {
    int s = i >> 6, d = i & 63;
    float acc = 0.f;
    for (int e = 0; e < D_; ++e) acc += sl[s][e] * Wq[d * D_ + e];
    qs[s][d] = acc;
  }
  __syncthreads();
  const float scale = 0.125f;  // 64^-0.5
  for (int i = t; i < S_ * D_; i += 256) {
    int s = i >> 6, e = i & 63;
    float acc = 0.f;
    for (int d = 0; d < D_; ++d) acc += qs[s][d] * Wk[d * D_ + e];
    qk[b * S_ * D_ + s * D_ + e] = scale * acc;
  }
}

// ---------------------------------------------------------------------------
// K2 (per iteration): single fused pass over x, double-buffered async staging.
//   per 16-row tile (one wave):
//     tile (4KB, contiguous) streamed via 8x global_load_async_to_lds_b128,
//       landing de-swizzled into a stride-68 LDS tile (conflict-free);
//       next tile's copy overlaps this tile's compute; s_wait_asynccnt syncs.
//     xln = LN(x)                         -> LDS reads -> registers (A-frag layout),
//                                            normalized values written back in place
//     logits[16x16] = xln @ qk^T          -> 16x V_WMMA_F32_16X16X4_F32
//     attn = softmax_s(logits)+eps        -> shfl_xor reductions over lane groups
//     AX  += attn^T @ xln                 -> 16x V_WMMA_F32_16X16X4_F32 (C persistent)
//   flush AX[7,64], CS[7] via global_atomic_add_f32.
//   NOTE: tile pointers are always derived directly from the __shared__ array
//   (never through a pointer stored in memory) so InferAddressSpaces keeps all
//   tile accesses on the DS path instead of FLAT.
// ---------------------------------------------------------------------------
__global__ __launch_bounds__(128) void k_attn(const float* __restrict__ x,
                                              const float* __restrict__ g, const float* __restrict__ bb,
                                              const float* __restrict__ qk,
                                              float* __restrict__ US, float* __restrict__ CS) {
  __shared__ float gs[D_], bs[D_];
  __shared__ float xt[WPB_][2][16 * 68];  // per-wave double buffer, stride 68 floats
  int tid = threadIdx.x;
  if (tid < 64) gs[tid] = g[tid];
  else if (tid < 128) bs[tid - 64] = bb[tid - 64];
  __syncthreads();

  int wave = tid >> 5, lane = tid & 31;
  int li = lane & 15, h = lane >> 4;
  int gw = blockIdx.x * WPB_ + wave;
  int tile0 = gw * TPW_;
  int batch = tile0 / TPB_;  // 2400 % 8 == 0 -> a wave never crosses batches

  // Preload B-fragments of qk^T (K=e, N=s): lane holds qk[s=li, k0+2h], k0=4t.
  v2f qkf[16];
  {
    const float* qb = qk + batch * S_ * D_ + li * D_;
#pragma unroll
    for (int t = 0; t < 16; ++t) {
      int k0 = 4 * t + 2 * h;
      if (li < S_) { qkf[t].x = qb[k0]; qkf[t].y = qb[k0 + 1]; }
      else { qkf[t].x = 0.f; qkf[t].y = 0.f; }
    }
  }

  v8f AXc[4] = {};   // persistent C fragments: AX[s (M), e (N: 4 tiles of 16)]
  float cs_loc = 0.f;

  // 32-bit LDS byte addresses for the async engine (low 32 bits of flat addr);
  // integers only -- no pointer provenance required for the asm path.
  unsigned bufb0 = (unsigned)(size_t)(void*)&xt[wave][0][0];
  unsigned bufb1 = (unsigned)(size_t)(void*)&xt[wave][1][0];
  // Per-lane invariant parts: global row = 2j+h, 16B column chunk = li.
  unsigned lane_lds = (unsigned)(h * 272 + li * 16);              // bytes into buffer
  unsigned long long lane_glb = (unsigned long long)(size_t)x +  // bytes into x
                                (unsigned long long)h * 256ull + (unsigned long long)li * 16ull;

  // Issue async copy of one 4KB tile into buffer base `lb0`.
  auto issue_tile = [&](int tile, unsigned lb0) {
    unsigned long long gb = lane_glb + (unsigned long long)tile * 4096ull;
    unsigned lb = lb0 + lane_lds;
#pragma unroll
    for (int j = 0; j < 8; ++j) {
      unsigned l = lb + (unsigned)(j * 544);           // (2j)*68*4 bytes
      unsigned long long gadr = gb + (unsigned long long)(j * 512);
      asm volatile("global_load_async_to_lds_b128 %0, %1, off"
                   :: "v"(l), "v"(gadr)
                   : "memory");
    }
  };

  issue_tile(tile0, bufb0);

  int aoff = li * 68 + 2 * h;                           // A-frag pattern offset (floats)

  for (int it = 0; it < TPW_; ++it) {
    async_wait0();                                      // tile `it` resident in buf[it&1]
    if (it + 1 < TPW_) issue_tile(tile0 + it + 1, (it & 1) ? bufb0 : bufb1);

    // Derive tile pointers straight from the shared array (keeps DS path).
    float* myx = &xt[wave][it & 1][0];
    float* rp = myx + aoff;                             // row li, elems 4t+2h

    // LayerNorm: partner lanes (l, l^16) hold complementary halves of each row.
    v2f xv[16];
#pragma unroll
    for (int t = 0; t < 16; ++t) xv[t] = *(const v2f*)(rp + 4 * t);

    float s0 = 0.f, s1 = 0.f;
#pragma unroll
    for (int t = 0; t < 16; ++t) { s0 += xv[t].x + xv[t].y; s1 += xv[t].x * xv[t].x + xv[t].y * xv[t].y; }
    s0 += __shfl_xor(s0, 16, 32);
    s1 += __shfl_xor(s1, 16, 32);
    float mean = s0 * (1.f / 64.f);
    float var = s1 * (1.f / 64.f) - mean * mean;
    float rs = rsqrtf(var + LN_EPS);

#pragma unroll
    for (int t = 0; t < 16; ++t) {
      int k0 = 4 * t + 2 * h;
      v2f o;
      o.x = (xv[t].x - mean) * rs * gs[k0] + bs[k0];
      o.y = (xv[t].y - mean) * rs * gs[k0 + 1] + bs[k0 + 1];
      xv[t] = o;
      *(v2f*)(rp + 4 * t) = o;  // write xln in place (lane-disjoint) for B-frag reads
    }

    // logits = xln @ qk^T  (A from registers, B preloaded, K=64 -> 16 WMMA steps)
    v8f lg = {0.f, 0.f, 0.f, 0.f, 0.f, 0.f, 0.f, 0.f};
#pragma unroll
    for (int t = 0; t < 16; ++t)
      lg = __builtin_amdgcn_wmma_f32_16x16x4_f32(false, xv[t], false, qkf[t], (short)0, lg, false, false);

    // Softmax over slots: C layout puts s on lanes (li), rows M = h*8+r in vgpr r.
    float attnr[8];
#pragma unroll
    for (int r = 0; r < 8; ++r) {
      float v = lg[r];
      float mx = (li < S_) ? v : -3.0e38f;
      mx = fmaxf(mx, __shfl_xor(mx, 1, 32));
      mx = fmaxf(mx, __shfl_xor(mx, 2, 32));
      mx = fmaxf(mx, __shfl_xor(mx, 4, 32));
      mx = fmaxf(mx, __shfl_xor(mx, 8, 32));
      float e = (li < S_) ? __expf(v - mx) : 0.f;
      float sm = e;
      sm += __shfl_xor(sm, 1, 32);
      sm += __shfl_xor(sm, 2, 32);
      sm += __shfl_xor(sm, 4, 32);
      sm += __shfl_xor(sm, 8, 32);
      float a = (li < S_) ? (e / sm + EPS_) : 0.f;
      attnr[r] = a;
      cs_loc += a;
    }

    // AX += attn^T @ xln : A = attn^T (M=s, K=row), B = xln (K=row, N=e).
#pragma unroll
    for (int t2 = 0; t2 < 4; ++t2) {
      int base = (t2 & 1) * 4;
      v2f af;
      if (t2 < 2) {  // needed rows live in lane half 0
        float c0 = __shfl_xor(attnr[base + 2], 16, 32);
        float c1 = __shfl_xor(attnr[base + 3], 16, 32);
        af.x = h ? c0 : attnr[base];
        af.y = h ? c1 : attnr[base + 1];
      } else {       // needed rows live in lane half 1
        float c0 = __shfl_xor(attnr[base], 16, 32);
        float c1 = __shfl_xor(attnr[base + 1], 16, 32);
        af.x = h ? attnr[base + 2] : c0;
        af.y = h ? attnr[base + 3] : c1;
      }
      int ra = 4 * t2 + 2 * h;
#pragma unroll
      for (int c = 0; c < 4; ++c) {
        v2f bf;
        bf.x = myx[ra * 68 + c * 16 + li];
        bf.y = myx[(ra + 1) * 68 + c * 16 + li];
        AXc[c] = __builtin_amdgcn_wmma_f32_16x16x4_f32(false, af, false, bf, (short)0, AXc[c], false, false);
      }
    }
  }

  // Flush per-wave partials.
  float cs_tot = cs_loc + __shfl_xor(cs_loc, 16, 32);
  if (h == 0 && li < S_) atomicAdd(&CS[batch * S_ + li], cs_tot);
  if (h == 0) {
#pragma unroll
    for (int c = 0; c < 4; ++c)
#pragma unroll
      for (int r = 0; r < S_; ++r)
        atomicAdd(&US[batch * S_ * D_ + r * D_ + c * 16 + li], AXc[c][r]);
  }
}

// ---------------------------------------------------------------------------
// K3 (per iteration): updates = (AX @ Wv^T)/CS ; GRU ; residual MLP. 112 rows.
// ---------------------------------------------------------------------------
__global__ __launch_bounds__(256) void k_finish(const float* __restrict__ US, const float* __restrict__ CS,
                                                const float* __restrict__ slots_in,
                                                const float* __restrict__ Wv,
                                                const float* __restrict__ Wih, const float* __restrict__ Whh,
                                                const float* __restrict__ bih, const float* __restrict__ bhh,
                                                const float* __restrict__ g, const float* __restrict__ bb,
                                                const float* __restrict__ W1, const float* __restrict__ b1v,
                                                const float* __restrict__ W2, const float* __restrict__ b2v,
                                                float* __restrict__ slots_out) {
  __shared__ float ax[S_][D_], sp[S_][D_], up[S_][D_], sn[S_][D_], ml[S_][D_], hh[S_][H_];
  __shared__ float mrs[S_][2], csh[S_];
  int b = blockIdx.x, t = threadIdx.x;
  for (int i = t; i < S_ * D_; i += 256) {
    ax[i >> 6][i & 63] = US[b * S_ * D_ + i];
    sp[i >> 6][i & 63] = slots_in[b * S_ * D_ + i];
  }
  if (t < S_) csh[t] = CS[b * S_ + t];
  __syncthreads();
  for (int i = t; i < S_ * D_; i += 256) {
    int s = i >> 6, d = i & 63;
    float acc = 0.f;
    for (int e = 0; e < D_; ++e) acc += ax[s][e] * Wv[d * D_ + e];
    up[s][d] = acc / csh[s];
  }
  __syncthreads();
  for (int i = t; i < S_ * D_; i += 256) {
    int s = i >> 6, d = i & 63;
    float ir = bih[d], iz = bih[64 + d], inn = bih[128 + d];
    float hr = bhh[d], hz = bhh[64 + d], hn = bhh[128 + d];
    for (int e = 0; e < D_; ++e) {
      float u = up[s][e], p = sp[s][e];
      ir += u * Wih[d * D_ + e];
      iz += u * Wih[(64 + d) * D_ + e];
      inn += u * Wih[(128 + d) * D_ + e];
      hr += p * Whh[d * D_ + e];
      hz += p * Whh[(64 + d) * D_ + e];
      hn += p * Whh[(128 + d) * D_ + e];
    }
    float r = sigmoidf_(ir + hr);
    float z = sigmoidf_(iz + hz);
    float n = tanhf(inn + r * hn);
    sn[s][d] = (1.f - z) * n + z * sp[s][d];
  }
  __syncthreads();
  if (t < S_) {
    float s0 = 0.f, s1 = 0.f;
    for (int d = 0; d < D_; ++d) { float v = sn[t][d]; s0 += v; s1 += v * v; }
    float m = s0 * (1.f / 64.f);
    float v = s1 * (1.f / 64.f) - m * m;
    mrs[t][0] = m; mrs[t][1] = rsqrtf(v + LN_EPS);
  }
  __syncthreads();
  for (int i = t; i < S_ * D_; i += 256) {
    int s = i >> 6, d = i & 63;
    ml[s][d] = (sn[s][d] - mrs[s][0]) * mrs[s][1] * g[d] + bb[d];
  }
  __syncthreads();
  for (int i = t; i < S_ * H_; i += 256) {
    int s = i / H_, j = i % H_;
    float acc = b1v[j];
    for (int d = 0; d < D_; ++d) acc += ml[s][d] * W1[j * D_ + d];
    hh[s][j] = fmaxf(acc, 0.f);
  }
  __syncthreads();
  for (int i = t; i < S_ * D_; i += 256) {
    int s = i >> 6, d = i & 63;
    float acc = b2v[d];
    for (int j = 0; j < H_; ++j) acc += hh[s][j] * W2[d * H_ + j];
    slots_out[b * S_ * D_ + i] = sn[s][d] + acc;
  }
}

// ---------------------------------------------------------------------------
extern "C" void kernel_launch(void* const* d_in, const int* in_sizes, int n_in,
                              void* d_out, int out_size, void* d_ws, size_t ws_size,
                              hipStream_t stream) {
  const float* x = (const float*)d_in[0];
  const float* sinit = (const float*)d_in[1];
  const float* mu = (const float*)d_in[2];
  const float* lsig = (const float*)d_in[3];
  const float* ln_in_g = (const float*)d_in[4];
  const float* ln_in_b = (const float*)d_in[5];
  const float* ln_sl_g = (const float*)d_in[6];
  const float* ln_sl_b = (const float*)d_in[7];
  const float* ln_ml_g = (const float*)d_in[8];
  const float* ln_ml_b = (const float*)d_in[9];
  const float* Wq = (const float*)d_in[10];
  const float* Wk = (const float*)d_in[11];
  const float* Wv = (const float*)d_in[12];
  const float* Wih = (const float*)d_in[13];
  const float* Whh = (const float*)d_in[14];
  const float* bih = (const float*)d_in[15];
  const float* bhh = (const float*)d_in[16];
  const float* W1 = (const float*)d_in[17];
  const float* b1 = (const float*)d_in[18];
  const float* W2 = (const float*)d_in[19];
  const float* b2 = (const float*)d_in[20];

  float* ws = (float*)d_ws;
  float* slots = ws;              // 7168 floats
  float* qk = ws + 7168;          // 7168
  float* US = ws + 2 * 7168;      // 7168
  float* CS = ws + 3 * 7168;      // 112
  float* out = (float*)d_out;

  k_init_slots<<<28, 256, 0, stream>>>(sinit, mu, lsig, slots);
  for (int iter = 0; iter < 3; ++iter) {
    k_qk<<<16, 256, 0, stream>>>(slots, ln_sl_g, ln_sl_b, Wq, Wk, qk, US, CS);
    k_attn<<<GRID2_, 128, 0, stream>>>(x, ln_in_g, ln_in_b, qk, US, CS);
    k_finish<<<16, 256, 0, stream>>>(US, CS, slots, Wv, Wih, Whh, bih, bhh,
                                     ln_ml_g, ln_ml_b, W1, b1, W2, b2,
                                     iter == 2 ? out : slots);
  }
}